// RDM_1219770712722
// MI455X (gfx1250) — compile-verified
//
#include <hip/hip_runtime.h>

typedef __attribute__((ext_vector_type(2))) float v2f;
typedef __attribute__((ext_vector_type(4))) float v4f;
typedef __attribute__((ext_vector_type(8))) float v8f;

#define B_   8
#define CL   64
#define HL   256
#define WL   256
#define CH   128
#define HH   64
#define WH   64
#define NLOW 4096   // HH*WH
#define RDM  64
#define KCH  8      // K-chunks for the cos GEMM (4096/8 = 512 per chunk)

__device__ __forceinline__ float sigmoidf_(float x) {
  return 1.0f / (1.0f + __expf(-x));
}

__device__ __forceinline__ float block_reduce_sum256(float v, float* sm) {
  int t = threadIdx.x;
  sm[t] = v;
  __syncthreads();
  for (int s = 128; s > 0; s >>= 1) {
    if (t < s) sm[t] += sm[t + s];
    __syncthreads();
  }
  float r = sm[0];
  __syncthreads();
  return r;
}

// K1: per (b,c_low) plane: bilinear align_corners downsample 256->64 into xl,
// plus full-plane mean (channel gate) and ||xl|| (cosine norm).
__global__ __launch_bounds__(256) void k_prep_low(const float* __restrict__ xlow,
    float* __restrict__ xl, float* __restrict__ nl, float* __restrict__ meanl) {
  __shared__ float sm[256];
  int bc = blockIdx.x;
  const float* src = xlow + (size_t)bc * (HL * WL);
  int t = threadIdx.x;
  // vectorized full-plane sum (b128 traffic)
  float s = 0.f;
  const v4f* src4 = (const v4f*)src;
  for (int i = t; i < (HL * WL) / 4; i += 256) {
    v4f v = src4[i];
    s += (v[0] + v[1]) + (v[2] + v[3]);
  }
  // downsample taps (L0/L2 hits after the streaming pass)
  float sq = 0.f;
  const float step = 255.0f / 63.0f;
  for (int o = t; o < NLOW; o += 256) {
    int oy = o >> 6, ox = o & 63;
    float py = oy * step; int iy0 = (int)py; float fy = py - iy0; int iy1 = min(iy0 + 1, HL - 1);
    float px = ox * step; int ix0 = (int)px; float fx = px - ix0; int ix1 = min(ix0 + 1, WL - 1);
    float v00 = src[iy0 * WL + ix0], v01 = src[iy0 * WL + ix1];
    float v10 = src[iy1 * WL + ix0], v11 = src[iy1 * WL + ix1];
    float v = (1.f - fy) * ((1.f - fx) * v00 + fx * v01) + fy * ((1.f - fx) * v10 + fx * v11);
    xl[(size_t)bc * NLOW + o] = v;
    sq += v * v;
  }
  float stot = block_reduce_sum256(s, sm);
  float sqtot = block_reduce_sum256(sq, sm);
  if (t == 0) {
    meanl[bc] = stot * (1.0f / (HL * WL));
    nl[bc] = sqrtf(sqtot) + 1e-12f;
  }
}

// K2: ||xh|| per (b, c_high)
__global__ __launch_bounds__(256) void k_prep_high(const float* __restrict__ xhigh,
    float* __restrict__ nh) {
  __shared__ float sm[256];
  const v4f* src4 = (const v4f*)(xhigh + (size_t)blockIdx.x * NLOW);
  float sq = 0.f;
  for (int i = threadIdx.x; i < NLOW / 4; i += 256) {
    v4f v = src4[i];
    sq += v[0] * v[0] + v[1] * v[1] + v[2] * v[2] + v[3] * v[3];
  }
  float tot = block_reduce_sum256(sq, sm);
  if (threadIdx.x == 0) nh[blockIdx.x] = sqrtf(tot) + 1e-12f;
}

// K3: raw partial dot products for cos: part[kc,b,c,d] = sum_{k in chunk} xl*xh.
// 64 blocks (8 batches x 8 K-chunks), 8 waves each: wave = 1 M-tile x 4 N-tiles.
// v_wmma_f32_16x16x4_f32, K = 512 per block for 8x better occupancy/latency hiding.
__global__ __launch_bounds__(256) void k_cos_gemm(const float* __restrict__ xl,
    const float* __restrict__ xhigh, float* __restrict__ part) {
  int b  = blockIdx.x >> 3;
  int kc = blockIdx.x & 7;
  int lane = threadIdx.x & 31;
  int wave = threadIdx.x >> 5;
  int mtile = wave >> 1;   // 0..3
  int nhalf = wave & 1;    // 0..1
  int row = mtile * 16 + (lane & 15);
  int kk = (lane < 16) ? 0 : 2;   // f32 A-frag: lanes 0-15 -> K 0,1 ; lanes 16-31 -> K 2,3
  const float* arow = xl + ((size_t)b * CL + row) * NLOW;
  const float* brow[4];
#pragma unroll
  for (int nt = 0; nt < 4; ++nt) {
    int col = (nhalf * 4 + nt) * 16 + (lane & 15);
    brow[nt] = xhigh + ((size_t)b * CH + col) * NLOW;
  }
  v8f acc[4] = {};
  int k0 = kc * (NLOW / KCH), k1 = k0 + (NLOW / KCH);
  for (int k = k0; k < k1; k += 4) {
    v2f a = *(const v2f*)(arow + k + kk);
#pragma unroll
    for (int nt = 0; nt < 4; ++nt) {
      v2f bf = *(const v2f*)(brow[nt] + k + kk);
      acc[nt] = __builtin_amdgcn_wmma_f32_16x16x4_f32(
          false, a, false, bf, (short)0, acc[nt], false, false);
    }
  }
  int mbase = mtile * 16 + ((lane >= 16) ? 8 : 0);
  int nlane = lane & 15;
#pragma unroll
  for (int nt = 0; nt < 4; ++nt) {
    int n = (nhalf * 4 + nt) * 16 + nlane;
#pragma unroll
    for (int r = 0; r < 8; ++r) {
      int m = mbase + r;
      part[(((size_t)kc * B_ + b) * CL + m) * CH + n] = acc[nt][r];
    }
  }
}

// K4: fold K-chunk partials + 1/(nl*nh) normalization + softmax over d (=128).
__global__ __launch_bounds__(128) void k_softmax(const float* __restrict__ part,
    const float* __restrict__ nl, const float* __restrict__ nh,
    float* __restrict__ cosb) {
  __shared__ float sm[128];
  int rowIdx = blockIdx.x;          // b*64 + m
  int b = rowIdx >> 6, m = rowIdx & 63;
  int t = threadIdx.x;              // d
  float v = 0.f;
#pragma unroll
  for (int kc = 0; kc < KCH; ++kc)
    v += part[(((size_t)kc * B_ + b) * CL + m) * CH + t];
  v /= (nl[b * CL + m] * nh[b * CH + t]);
  sm[t] = v; __syncthreads();
  for (int s = 64; s > 0; s >>= 1) { if (t < s) sm[t] = fmaxf(sm[t], sm[t + s]); __syncthreads(); }
  float mx = sm[0]; __syncthreads();
  float e = __expf(v - mx);
  sm[t] = e; __syncthreads();
  for (int s = 64; s > 0; s >>= 1) { if (t < s) sm[t] += sm[t + s]; __syncthreads(); }
  cosb[(size_t)rowIdx * CH + t] = e / sm[0];
}

// K5: dual GEMM over shared B operand xh (K=128):
//   rows 0..63:  xhc = cos @ xh     rows 64..127: red64 = relu(bn(red_w @ xh))
__global__ __launch_bounds__(256) void k_apply_gemm(const float* __restrict__ cosb,
    const float* __restrict__ red_w, const float* __restrict__ xhigh,
    const float* __restrict__ red_gamma, const float* __restrict__ red_beta,
    const float* __restrict__ red_mean, const float* __restrict__ red_var,
    float* __restrict__ xhc, float* __restrict__ red64) {
  int b = blockIdx.x >> 8;
  int lane = threadIdx.x & 31;
  int wave = threadIdx.x >> 5;
  int tileIdx = (blockIdx.x & 255) * 8 + wave;  // 0..2047 per batch
  int mtile = tileIdx >> 8;                     // 0..7 (0-3: cos rows, 4-7: red rows)
  int ntile = tileIdx & 255;                    // 0..255
  int kk = (lane < 16) ? 0 : 2;
  int row = (mtile & 3) * 16 + (lane & 15);
  const float* arow = (mtile < 4) ? (cosb + ((size_t)b * CL + row) * CH)
                                  : (red_w + (size_t)row * CH);
  int col = ntile * 16 + (lane & 15);
  const float* bbase = xhigh + (size_t)b * CH * NLOW + col;
  v8f acc = {};
  for (int k4 = 0; k4 < CH; k4 += 4) {
    v2f a = *(const v2f*)(arow + k4 + kk);
    int d0 = k4 + kk;
    v2f bf;
    bf[0] = bbase[(size_t)d0 * NLOW];
    bf[1] = bbase[(size_t)(d0 + 1) * NLOW];
    acc = __builtin_amdgcn_wmma_f32_16x16x4_f32(
        false, a, false, bf, (short)0, acc, false, false);
  }
  int mbase = (mtile & 3) * 16 + ((lane >= 16) ? 8 : 0);
  if (mtile < 4) {
#pragma unroll
    for (int r = 0; r < 8; ++r) {
      int m = mbase + r;
      xhc[((size_t)b * CL + m) * NLOW + col] = acc[r];
    }
  } else {
#pragma unroll
    for (int r = 0; r < 8; ++r) {
      int o = mbase + r;
      float inv = rsqrtf(red_var[o] + 1e-5f);
      float val = (acc[r] - red_mean[o]) * (red_gamma[o] * inv) + red_beta[o];
      red64[((size_t)b * RDM + o) * NLOW + col] = fmaxf(val, 0.f);
    }
  }
}

// K6a: exact mean of align_corners-upsampled red64 via separable
// interpolation-weight vector (built with LDS float atomics -> ds_add_f32).
__global__ __launch_bounds__(256) void k_red_up_mean(const float* __restrict__ red64,
    float* __restrict__ rum) {
  __shared__ float w[64];
  __shared__ float sm[256];
  int t = threadIdx.x;
  if (t < 64) w[t] = 0.f;
  __syncthreads();
  {
    float p = t * (63.0f / 255.0f);
    int i0 = (int)p; float f = p - i0; int i1 = min(i0 + 1, 63);
    atomicAdd(&w[i0], 1.0f - f);
    atomicAdd(&w[i1], f);
  }
  __syncthreads();
  const float* src = red64 + (size_t)blockIdx.x * NLOW;
  float s = 0.f;
  for (int i = t; i < NLOW; i += 256) s += w[i >> 6] * w[i & 63] * src[i];
  float tot = block_reduce_sum256(s, sm);
  if (t == 0) rum[blockIdx.x] = tot * (1.0f / (HL * WL));
}

// K6b: channel gate: att = sigmoid(bn(cconv_w @ pooled + b))
__global__ __launch_bounds__(64) void k_att(const float* __restrict__ meanl,
    const float* __restrict__ rum, const float* __restrict__ cconv_w,
    const float* __restrict__ cconv_b, const float* __restrict__ cc_gamma,
    const float* __restrict__ cc_beta, const float* __restrict__ cc_mean,
    const float* __restrict__ cc_var, float* __restrict__ att) {
  __shared__ float pooled[2 * RDM];
  int b = blockIdx.x;
  int o = threadIdx.x;
  pooled[o] = meanl[b * CL + o];
  pooled[RDM + o] = rum[b * RDM + o];
  __syncthreads();
  float acc = cconv_b[o];
#pragma unroll 4
  for (int c = 0; c < 2 * RDM; ++c) acc += cconv_w[o * 2 * RDM + c] * pooled[c];
  float inv = rsqrtf(cc_var[o] + 1e-5f);
  float v = (acc - cc_mean[o]) * (cc_gamma[o] * inv) + cc_beta[o];
  att[b * RDM + o] = sigmoidf_(v);
}

// K7: fused upsample(64->256, align_corners) + gate + sigmoid-diff + ReLU.
// Sources staged memory->LDS with CDNA5 async copies (ASYNCcnt-tracked, no
// VGPR round trip); output streamed with non-temporal b128 stores so the
// 268 MB result doesn't evict x_low (L2-resident between K1 and K7).
__global__ __launch_bounds__(256) void k_final(const float* __restrict__ xlow,
    const float* __restrict__ xhc, const float* __restrict__ red64,
    const float* __restrict__ att, float* __restrict__ out) {
  __shared__ float sA[NLOW];
  __shared__ float sR[NLOW];
  int bc = blockIdx.x;
  int b = bc >> 6;
  int c = bc & 63;
  int t = threadIdx.x;
  const float* xlp = xlow + (size_t)bc * (HL * WL);
  const float* xap = xhc + (size_t)bc * NLOW;
  const float* xrp = red64 + (size_t)bc * NLOW;
  // async memory->LDS staging: 16B chunks, 1024 per plane, 4 per thread
  unsigned ldsA = (unsigned)(size_t)(void*)sA;   // flat-aperture low 32 bits == LDS addr
  unsigned ldsR = (unsigned)(size_t)(void*)sR;
  for (int i = t; i < NLOW / 4; i += 256) {
    unsigned la = ldsA + 16u * i;
    unsigned lr = ldsR + 16u * i;
    unsigned long long ga = (unsigned long long)(size_t)(xap + 4 * i);
    unsigned long long gr = (unsigned long long)(size_t)(xrp + 4 * i);
    asm volatile("global_load_async_to_lds_b128 %0, %1, off"
                 :: "v"(la), "v"(ga) : "memory");
    asm volatile("global_load_async_to_lds_b128 %0, %1, off"
                 :: "v"(lr), "v"(gr) : "memory");
  }
  float gate = att[bc];
  asm volatile("s_wait_asynccnt 0x0" ::: "memory");
  __syncthreads();
  float* out0 = out + ((size_t)b * CH + c) * (HL * WL);
  float* out1 = out + ((size_t)b * CH + RDM + c) * (HL * WL);
  const float step = 63.0f / 255.0f;
  for (int g = t; g < (HL * WL) / 4; g += 256) {
    int y = g >> 6;
    int x4 = (g & 63) * 4;
    float py = y * step; int iy0 = (int)py; float fy = py - iy0; int iy1 = min(iy0 + 1, HH - 1);
    int rb0 = iy0 * WH, rb1 = iy1 * WH;
    v4f xin = *(const v4f*)(xlp + y * WL + x4);
    v4f r0, r1;
#pragma unroll
    for (int j = 0; j < 4; ++j) {
      int x = x4 + j;
      float px = x * step; int ix0 = (int)px; float fx = px - ix0; int ix1 = min(ix0 + 1, WH - 1);
      float w00 = (1.f - fy) * (1.f - fx), w01 = (1.f - fy) * fx;
      float w10 = fy * (1.f - fx),         w11 = fy * fx;
      float vA = w00 * sA[rb0 + ix0] + w01 * sA[rb0 + ix1] + w10 * sA[rb1 + ix0] + w11 * sA[rb1 + ix1];
      float vR = (w00 * sR[rb0 + ix0] + w01 * sR[rb0 + ix1] + w10 * sR[rb1 + ix0] + w11 * sR[rb1 + ix1]) * gate;
      float sl = sigmoidf_(xin[j]);
      r0[j] = fmaxf(sl - sigmoidf_(vA), 0.f);
      r1[j] = fmaxf(sl - sigmoidf_(vR), 0.f);
    }
    __builtin_nontemporal_store(r0, (v4f*)(out0 + y * WL + x4));
    __builtin_nontemporal_store(r1, (v4f*)(out1 + y * WL + x4));
  }
}

extern "C" void kernel_launch(void* const* d_in, const int* in_sizes, int n_in,
                              void* d_out, int out_size, void* d_ws, size_t ws_size,
                              hipStream_t stream) {
  (void)in_sizes; (void)n_in; (void)out_size; (void)ws_size;
  const float* x_low     = (const float*)d_in[0];
  const float* x_high    = (const float*)d_in[1];
  const float* red_w     = (const float*)d_in[2];
  const float* red_gamma = (const float*)d_in[3];
  const float* red_beta  = (const float*)d_in[4];
  const float* red_mean  = (const float*)d_in[5];
  const float* red_var   = (const float*)d_in[6];
  const float* cconv_w   = (const float*)d_in[7];
  const float* cconv_b   = (const float*)d_in[8];
  const float* cc_gamma  = (const float*)d_in[9];
  const float* cc_beta   = (const float*)d_in[10];
  const float* cc_mean   = (const float*)d_in[11];
  const float* cc_var    = (const float*)d_in[12];
  float* out = (float*)d_out;
  float* ws = (float*)d_ws;

  float* xl    = ws;                  // 8*64*4096        = 2097152
  float* xhc   = xl + 2097152;        // 8*64*4096
  float* red64 = xhc + 2097152;       // 8*64*4096
  float* part  = red64 + 2097152;     // KCH*8*64*128     = 524288
  float* cosb  = part + 524288;       // 8*64*128         = 65536
  float* nl    = cosb + 65536;        // 512
  float* nh    = nl + 512;            // 1024
  float* meanl = nh + 1024;           // 512
  float* rum   = meanl + 512;         // 512
  float* attb  = rum + 512;           // 512

  k_prep_low<<<B_ * CL, 256, 0, stream>>>(x_low, xl, nl, meanl);
  k_prep_high<<<B_ * CH, 256, 0, stream>>>(x_high, nh);
  k_cos_gemm<<<B_ * KCH, 256, 0, stream>>>(xl, x_high, part);
  k_softmax<<<B_ * CL, 128, 0, stream>>>(part, nl, nh, cosb);
  k_apply_gemm<<<B_ * 256, 256, 0, stream>>>(cosb, red_w, x_high,
      red_gamma, red_beta, red_mean, red_var, xhc, red64);
  k_red_up_mean<<<B_ * RDM, 256, 0, stream>>>(red64, rum);
  k_att<<<B_, 64, 0, stream>>>(meanl, rum, cconv_w, cconv_b,
      cc_gamma, cc_beta, cc_mean, cc_var, attb);
  k_final<<<B_ * CL, 256, 0, stream>>>(x_low, xhc, red64, attb, out);
}